// GraphSAGE_20512763806336
// MI455X (gfx1250) — compile-verified
//
#include <hip/hip_runtime.h>
#include <hip/hip_bf16.h>

typedef float v2f __attribute__((ext_vector_type(2)));
typedef float v8f __attribute__((ext_vector_type(8)));

#define DIM_K 128

// ---------------------------------------------------------------------------
// zero-fill (float4 granularity)
// ---------------------------------------------------------------------------
__global__ void zero4_kernel(float4* __restrict__ p, int n4) {
    int i = blockIdx.x * blockDim.x + threadIdx.x;
    if (i < n4) p[i] = make_float4(0.f, 0.f, 0.f, 0.f);
}

// ---------------------------------------------------------------------------
// edge scatter: one wave per edge; lane owns a float4 slice of the 128-dim row.
// Row read is 32 lanes x 16B = 512B fully coalesced; adds are L2-resident
// global_atomic_add_f32. Optionally accumulates the in-degree count.
// ---------------------------------------------------------------------------
__global__ void scatter_kernel(const float* __restrict__ feat,
                               const long long* __restrict__ esrc,
                               const long long* __restrict__ edst,
                               float* __restrict__ agg,
                               float* __restrict__ cnt,
                               int nEdges, int addCount) {
    int gw   = (blockIdx.x * blockDim.x + threadIdx.x) >> 5;
    int lane = threadIdx.x & 31;
    if (gw >= nEdges) return;
    int s = (int)esrc[gw];
    int d = (int)edst[gw];
    const float4* srow = (const float4*)(feat + (size_t)s * DIM_K);
    float4 v = srow[lane];
    float* dp = agg + (size_t)d * DIM_K + lane * 4;
    atomicAdd(dp + 0, v.x);
    atomicAdd(dp + 1, v.y);
    atomicAdd(dp + 2, v.z);
    atomicAdd(dp + 3, v.w);
    if (addCount && lane == 0) atomicAdd(cnt + d, 1.0f);
}

// ---------------------------------------------------------------------------
// SAGE layer: out[16-row tile] = relu( (agg/max(cnt,1)) @ Wl^T + x @ Wr^T + b )
// Block = 128 threads (4 waves), one 16-row M-tile per block (50000 = 3125*16,
// no divergence -> EXEC all-ones for WMMA). Wave owns NT = NOUT/64 column
// tiles, processed together inside the single k-loop so A-fragments are
// consumed immediately (no cross-loop hoisting -> no scratch spills).
// fp32 matrix math via V_WMMA_F32_16X16X4_F32, K=128 in 32 steps.
// ---------------------------------------------------------------------------
template <int NOUT>
__global__ void sage_kernel(const float* __restrict__ xin,
                            const float* __restrict__ agg,
                            const float* __restrict__ cnt,
                            const float* __restrict__ Wl,
                            const float* __restrict__ bias,
                            const float* __restrict__ Wr,
                            float* __restrict__ out) {
    // 132-float pitch: 132 % 64 = 4 -> conflict-free column reads; row pitch
    // is 528 B (8B aligned) so even-k v2f LDS loads are legal.
    __shared__ float xT[16][132];
    __shared__ float mT[16][132];

    const int tid  = threadIdx.x;
    const int row0 = blockIdx.x * 16;

    // ---- stage A-tiles: 8 threads per row, 16 cols each (4 x float4) ----
    {
        int r  = tid >> 3;
        int cs = (tid & 7) << 4;
        float c   = cnt[row0 + r];
        float inv = 1.0f / fmaxf(c, 1.0f);
        const float4* xr = (const float4*)(xin + (size_t)(row0 + r) * DIM_K + cs);
        const float4* ar = (const float4*)(agg + (size_t)(row0 + r) * DIM_K + cs);
#pragma unroll
        for (int j = 0; j < 4; ++j) {
            float4 xv = xr[j];
            float4 av = ar[j];
            *(float4*)&xT[r][cs + 4 * j] = xv;
            av.x *= inv; av.y *= inv; av.z *= inv; av.w *= inv;
            *(float4*)&mT[r][cs + 4 * j] = av;
        }
    }
    __syncthreads();

    constexpr int NT = NOUT / 64;       // column tiles per wave (2 or 1)

    const int wave = tid >> 5;
    const int lane = tid & 31;
    const int half = lane >> 4;   // 0: K={0,1} | 1: K={2,3}
    const int m    = lane & 15;   // A: M index, B: N index, C/D: N index
    const int kk   = half << 1;

    int nbase[NT];
    const float* wlrow[NT];
    const float* wrrow[NT];
    v8f acc[NT];
#pragma unroll
    for (int t = 0; t < NT; ++t) {
        nbase[t] = (wave + 4 * t) << 4;
        wlrow[t] = Wl + (size_t)(nbase[t] + m) * DIM_K + kk;  // B = Wl^T
        wrrow[t] = Wr + (size_t)(nbase[t] + m) * DIM_K + kk;  // B = Wr^T
        acc[t]   = (v8f){};
    }

#pragma unroll 4
    for (int k0 = 0; k0 < DIM_K; k0 += 4) {
        v2f am = *(const v2f*)&mT[m][k0 + kk];
        v2f ax = *(const v2f*)&xT[m][k0 + kk];
#pragma unroll
        for (int t = 0; t < NT; ++t) {
            v2f bl_ = *(const v2f*)(wlrow[t] + k0);
            v2f br_ = *(const v2f*)(wrrow[t] + k0);
            acc[t] = __builtin_amdgcn_wmma_f32_16x16x4_f32(
                false, am, false, bl_, (short)0, acc[t], false, false);
            acc[t] = __builtin_amdgcn_wmma_f32_16x16x4_f32(
                false, ax, false, br_, (short)0, acc[t], false, false);
        }
    }

#pragma unroll
    for (int t = 0; t < NT; ++t) {
        const float bv = bias[nbase[t] + m];
#pragma unroll
        for (int r2 = 0; r2 < 8; ++r2) {
            int row = row0 + (half << 3) + r2;  // C/D: M = 8*half + vgpr idx
            float v = acc[t][r2] + bv;
            out[(size_t)row * NOUT + nbase[t] + m] = v > 0.0f ? v : 0.0f;
        }
    }
}

// ---------------------------------------------------------------------------
// in-place log_softmax over 64 columns: one wave per row, 2 cols per lane,
// wave32 shuffle reductions.
// ---------------------------------------------------------------------------
__global__ void logsoftmax64_kernel(float* __restrict__ data, int nrows) {
    int wid  = (blockIdx.x * blockDim.x + threadIdx.x) >> 5;
    int lane = threadIdx.x & 31;
    if (wid >= nrows) return;
    float* row = data + (size_t)wid * 64;
    float v0 = row[lane];
    float v1 = row[lane + 32];
    float mx = fmaxf(v0, v1);
#pragma unroll
    for (int o = 16; o > 0; o >>= 1) mx = fmaxf(mx, __shfl_xor(mx, o, 32));
    float s = expf(v0 - mx) + expf(v1 - mx);
#pragma unroll
    for (int o = 16; o > 0; o >>= 1) s += __shfl_xor(s, o, 32);
    float ls = logf(s);
    row[lane]      = v0 - mx - ls;
    row[lane + 32] = v1 - mx - ls;
}

// ---------------------------------------------------------------------------
extern "C" void kernel_launch(void* const* d_in, const int* in_sizes, int n_in,
                              void* d_out, int out_size, void* d_ws, size_t ws_size,
                              hipStream_t stream) {
    const float*     x   = (const float*)d_in[0];
    const long long* ei  = (const long long*)d_in[1];
    const float*     Wl1 = (const float*)d_in[2];
    const float*     bl1 = (const float*)d_in[3];
    const float*     Wr1 = (const float*)d_in[4];
    const float*     Wl2 = (const float*)d_in[5];
    const float*     bl2 = (const float*)d_in[6];
    const float*     Wr2 = (const float*)d_in[7];

    const int NN = in_sizes[0] / DIM_K;   // 50000
    const int NE = in_sizes[1] / 2;       // 800000

    const long long* esrc = ei;
    const long long* edst = ei + NE;

    // workspace layout: agg[NN*128] | cnt[NN] | h[NN*128]
    float* agg = (float*)d_ws;
    float* cnt = agg + (size_t)NN * DIM_K;
    float* h   = cnt + NN;

    float* out = (float*)d_out;

    const int scatterBlocks = (NE * 32 + 255) / 256;  // one wave per edge
    const int tileBlocks    = NN / 16;                // 3125

    // ---- layer 1 ----
    {   // zero agg+cnt in one contiguous pass (NN*129 floats, mult of 4)
        int n4 = (NN * (DIM_K + 1)) / 4;
        zero4_kernel<<<(n4 + 255) / 256, 256, 0, stream>>>((float4*)agg, n4);
    }
    scatter_kernel<<<scatterBlocks, 256, 0, stream>>>(x, esrc, edst, agg, cnt, NE, 1);
    sage_kernel<128><<<tileBlocks, 128, 0, stream>>>(x, agg, cnt, Wl1, bl1, Wr1, h);

    // ---- layer 2 (counts unchanged; re-zero agg only) ----
    {
        int n4 = (NN * DIM_K) / 4;
        zero4_kernel<<<(n4 + 255) / 256, 256, 0, stream>>>((float4*)agg, n4);
    }
    scatter_kernel<<<scatterBlocks, 256, 0, stream>>>(h, esrc, edst, agg, cnt, NE, 0);
    sage_kernel<64><<<tileBlocks, 128, 0, stream>>>(h, agg, cnt, Wl2, bl2, Wr2, out);

    // ---- log_softmax in place on d_out ----
    {
        int threads = NN * 32;
        logsoftmax64_kernel<<<(threads + 255) / 256, 256, 0, stream>>>(out, NN);
    }
}